// ResidualVectorQuantizer_87230785782025
// MI455X (gfx1250) — compile-verified
//
#include <hip/hip_runtime.h>
#include <hip/hip_bf16.h>

// Residual VQ forward for MI455X (gfx1250, wave32, WMMA).
// z: [16,128,4500] f32, codebooks: [8,1024,128] f32.
// Outputs (concatenated in d_out, float32):
//   quantized [16,128,4500] (9,216,000), codes [8,16,4500] (576,000),
//   loss_commit (1), loss_codebook (1).

#define BB 16
#define DD 128
#define TT 4500
#define LL 8
#define KK 1024
#define NVEC (BB * TT)          // 72000
#define QUANT_OFF 0
#define CODES_OFF (BB * DD * TT)          // 9,216,000
#define LOSS_OFF (CODES_OFF + LL * NVEC)  // 9,792,000
#define INV_MEAN (1.0f / 9216000.0f)      // 1/(B*T*D)

typedef __attribute__((ext_vector_type(2))) float v2f;
typedef __attribute__((ext_vector_type(8))) float v8f;

// ---------------------------------------------------------------------------
// Kernel 1: codebook squared norms into d_ws, zero the loss accumulators.
// ---------------------------------------------------------------------------
__global__ void rvq_y2_kernel(const float* __restrict__ cb,
                              float* __restrict__ y2,
                              float* __restrict__ out) {
  int i = blockIdx.x * blockDim.x + threadIdx.x;
  if (i == 0) {
    out[LOSS_OFF + 0] = 0.0f;
    out[LOSS_OFF + 1] = 0.0f;
  }
  if (i < LL * KK) {
    const float* p = cb + (size_t)i * DD;
    float s = 0.0f;
#pragma unroll 8
    for (int d = 0; d < DD; ++d) s += p[d] * p[d];
    y2[i] = s;
  }
}

// ---------------------------------------------------------------------------
// Kernel 2: main RVQ. 128 threads = 4 waves; each wave owns one 16-vector tile.
// ---------------------------------------------------------------------------
__global__ void __launch_bounds__(128)
rvq_main_kernel(const float* __restrict__ z,
                const float* __restrict__ cb,
                const float* __restrict__ y2,
                float* __restrict__ out) {
  __shared__ float s_r[4 * DD * 16];   // residual tiles, [wave][d][row]
  __shared__ float s_zq[4 * DD * 16];  // z_q accumulators
  __shared__ int s_idx[4 * 16];        // chosen code per row per wave

  const int lane = threadIdx.x & 31;
  const int wave = threadIdx.x >> 5;
  const int row = lane & 15;   // M-row for A, N-col for B/D fragments
  const int hi = lane >> 4;    // lane half selects K sub-pair / M+8
  const int wbase = wave * (DD * 16);
  const int n_base = (blockIdx.x * 4 + wave) * 16;  // first flat vector id

  // ---- Stage residual tile into LDS (z is [B,D,T], row-of-16 along T) ----
  for (int i = lane; i < DD * 16; i += 32) {
    int r_ = i & 15;
    int d = i >> 4;
    unsigned nv = (unsigned)(n_base + r_);
    unsigned b = nv / TT;
    unsigned tt = nv - b * TT;
    s_r[wbase + i] = z[((size_t)b * DD + d) * TT + tt];
    s_zq[wbase + i] = 0.0f;
  }
  __syncthreads();

  float errAcc = 0.0f;

  for (int l = 0; l < LL; ++l) {
    const float* cbL = cb + (size_t)l * KK * DD;
    const float* y2L = y2 + l * KK;

    // ---- A fragments: 16x4 f32 per K-chunk; lane holds contiguous pair ----
    v2f a[32];
#pragma unroll
    for (int kc = 0; kc < 32; ++kc) {
      int dbase = 4 * kc + 2 * hi;
      a[kc].x = s_r[wbase + dbase * 16 + row];
      a[kc].y = s_r[wbase + (dbase + 1) * 16 + row];
    }

    float best_s[8];
    int best_i[8];
#pragma unroll
    for (int v = 0; v < 8; ++v) {
      best_s[v] = 3.4e38f;
      best_i[v] = 0;
    }

    // ---- Sweep 64 code tiles of 16 codes each ----
    for (int ct = 0; ct < 64; ++ct) {
      const int c0 = ct * 16;
      // B fragment base: this lane supplies code (c0+row), elems 4kc+2hi,+1
      const float* bbase = cbL + (size_t)(c0 + row) * DD + 2 * hi;
      if (ct < 63) __builtin_prefetch(bbase + 16 * DD, 0, 0);  // next tile

      // Two independent accumulators (even/odd K-chunks) break the
      // WMMA->WMMA RAW chain so consecutive wmma ops never stall.
      v8f acc0 = {};
      v8f acc1 = {};
#pragma unroll
      for (int kc = 0; kc < 16; ++kc) {
        v2f bf0 = *reinterpret_cast<const v2f*>(bbase + 8 * kc);
        v2f bf1 = *reinterpret_cast<const v2f*>(bbase + 8 * kc + 4);
        acc0 = __builtin_amdgcn_wmma_f32_16x16x4_f32(
            false, a[2 * kc], false, bf0, (short)0, acc0, false, false);
        acc1 = __builtin_amdgcn_wmma_f32_16x16x4_f32(
            false, a[2 * kc + 1], false, bf1, (short)0, acc1, false, false);
      }

      // score = ||c||^2 - 2*dot  (same argmin ordering as x2+y2-2xy)
      const float y2v = y2L[c0 + row];
      const int ci = c0 + row;
#pragma unroll
      for (int v = 0; v < 8; ++v) {
        float dot = acc0[v] + acc1[v];
        float s = __builtin_fmaf(-2.0f, dot, y2v);
        if (s < best_s[v]) {
          best_s[v] = s;
          best_i[v] = ci;
        }
      }
    }

    // ---- Argmin reduction across the 16 N-lanes of each half ----
#pragma unroll
    for (int v = 0; v < 8; ++v) {
      float s = best_s[v];
      int bi = best_i[v];
#pragma unroll
      for (int off = 8; off >= 1; off >>= 1) {
        float os = __shfl_xor(s, off, 32);
        int oi = __shfl_xor(bi, off, 32);
        if (os < s || (os == s && oi < bi)) {
          s = os;
          bi = oi;
        }
      }
      if (row == 0) {  // lanes 0 and 16 hold rows v and v+8
        int m = v + 8 * hi;
        s_idx[wave * 16 + m] = bi;
        out[CODES_OFF + (size_t)l * NVEC + n_base + m] = (float)bi;
      }
    }
    __syncthreads();

    // ---- Residual update, z_q accumulate, loss partial ----
    for (int i = lane; i < DD * 16; i += 32) {
      int r_ = i & 15;
      int d = i >> 4;
      float q = cbL[(size_t)s_idx[wave * 16 + r_] * DD + d];
      float rv = s_r[wbase + i];
      float diff = rv - q;
      errAcc = __builtin_fmaf(diff, diff, errAcc);
      s_zq[wbase + i] += q;
      s_r[wbase + i] = diff;
    }
    __syncthreads();
  }

  // ---- Write quantized output (back to [B,D,T]) ----
  for (int i = lane; i < DD * 16; i += 32) {
    int r_ = i & 15;
    int d = i >> 4;
    unsigned nv = (unsigned)(n_base + r_);
    unsigned b = nv / TT;
    unsigned tt = nv - b * TT;
    out[QUANT_OFF + ((size_t)b * DD + d) * TT + tt] = s_zq[wbase + i];
  }

  // ---- Loss reduction: wave-reduce then one atomic per wave per slot ----
#pragma unroll
  for (int off = 16; off >= 1; off >>= 1)
    errAcc += __shfl_xor(errAcc, off, 32);
  if (lane == 0) {
    float c = errAcc * INV_MEAN;
    atomicAdd(&out[LOSS_OFF + 0], c);  // loss_commit
    atomicAdd(&out[LOSS_OFF + 1], c);  // loss_codebook (== commit in eval)
  }
}

// ---------------------------------------------------------------------------
extern "C" void kernel_launch(void* const* d_in, const int* in_sizes, int n_in,
                              void* d_out, int out_size, void* d_ws,
                              size_t ws_size, hipStream_t stream) {
  const float* z = (const float*)d_in[0];         // [16,128,4500]
  const float* cb = (const float*)d_in[1];        // [8,1024,128]
  float* out = (float*)d_out;
  float* y2 = (float*)d_ws;                       // 8192 floats

  // Kernel 1: ||c||^2 table + zero loss slots (same stream => ordered).
  rvq_y2_kernel<<<(LL * KK + 127) / 128, 128, 0, stream>>>(cb, y2, out);

  // Kernel 2: 4500 tiles of 16 vectors, 4 tiles (waves) per block.
  rvq_main_kernel<<<NVEC / 64, 128, 0, stream>>>(z, cb, y2, out);
}